// LFD_37503654428951
// MI455X (gfx1250) — compile-verified
//
#include <hip/hip_runtime.h>
#include <stdint.h>

#define P_TOTAL 21824
#define G_GT    100
#define NCLS    80
#define BLK     64
#define CLS_STRIDE 81   // 81*tid mod 64 banks -> 17*tid mod 64, gcd(17,64)=1: conflict-free

typedef unsigned int uint32x4 __attribute__((ext_vector_type(4)));
typedef int          int32x8  __attribute__((ext_vector_type(8)));
typedef int          int32x4  __attribute__((ext_vector_type(4)));

// 1D dword copy global->LDS via the Tensor Data Mover (D# per CDNA5 ISA §8.3/8.4).
__device__ __forceinline__ void tdm_copy_dwords(unsigned lds_byte_addr,
                                                unsigned long long gaddr,
                                                unsigned n_dwords) {
  uint32x4 g0;
  g0.x = 1u;                                              // count=1, user descriptor
  g0.y = lds_byte_addr;                                   // lds_addr (bytes)
  g0.z = (unsigned)(gaddr & 0xFFFFFFFFull);               // global_addr[31:0]
  g0.w = (unsigned)((gaddr >> 32) & 0x01FFFFFFull)        // global_addr[56:32]
         | (2u << 30);                                    // type = 2 ("image")

  int32x8 g1;
  g1[0] = (int)(2u << 16);                                // data_size = 4 bytes
  g1[1] = (int)((n_dwords & 0xFFFFu) << 16);              // tensor_dim0[15:0] @ bits[63:48]
  g1[2] = (int)(((n_dwords >> 16) & 0xFFFFu)              // tensor_dim0[31:16]
                | (1u << 16));                            // tensor_dim1 = 1 (lo half)
  g1[3] = (int)((n_dwords & 0xFFFFu) << 16);              // tile_dim0 = n @ bits[127:112]
  g1[4] = 1;                                              // tile_dim1 = 1, tile_dim2 = 0
  g1[5] = (int)n_dwords;                                  // tensor_dim0_stride lo
  g1[6] = 0;
  g1[7] = 0;

  int32x4 gz4 = {0, 0, 0, 0};                             // groups 2/3 unused (<=2D)
  int32x8 gz8 = {0, 0, 0, 0, 0, 0, 0, 0};                 // extra group (6-arg toolchain)
  __builtin_amdgcn_tensor_load_to_lds(g0, g1, gz4, gz4, gz8, 0);
}

__global__ __launch_bounds__(BLK) void LFD_assign_kernel(
    const float* __restrict__ gt_bboxes,    // [B, G, 4]
    const int*   __restrict__ gt_labels,    // [B, G]
    const float* __restrict__ points,       // [P, 2]
    const float* __restrict__ reg_ranges,   // [P, 2]
    const float* __restrict__ gray_ranges,  // [P, 2]
    const float* __restrict__ strides,      // [P]
    float* __restrict__ out)                // [B, P, 84]
{
  __shared__ float s_bbox[G_GT * 4];
  __shared__ int   s_label[G_GT];
  __shared__ float s_cls[BLK * CLS_STRIDE];

  const int tid = threadIdx.x;
  const int b   = blockIdx.y;
  const int p   = blockIdx.x * BLK + tid;   // grid sized so p < P always

  // --- stage GT data into LDS with the Tensor Data Mover (one issue per block) ---
  if (tid == 0) {
    tdm_copy_dwords((unsigned)(uintptr_t)(&s_bbox[0]),
                    (unsigned long long)(uintptr_t)(gt_bboxes + (size_t)b * G_GT * 4),
                    G_GT * 4);
    tdm_copy_dwords((unsigned)(uintptr_t)(&s_label[0]),
                    (unsigned long long)(uintptr_t)(gt_labels + (size_t)b * G_GT),
                    G_GT);
  }
  __builtin_amdgcn_s_wait_tensorcnt(0);
  __syncthreads();

  // --- per-point constants ---
  const float px  = points[2 * p + 0];
  const float py  = points[2 * p + 1];
  const float lo  = reg_ranges[2 * p + 0];
  const float up  = reg_ranges[2 * p + 1];
  const float glo = gray_ranges[2 * p + 0];
  const float gup = gray_ranges[2 * p + 1];
  const float inv_half = 2.0f / strides[p];
  const float inv_up   = 1.0f / up;

  // zero per-thread class maxima
#pragma unroll
  for (int c = 0; c < NCLS; ++c) s_cls[tid * CLS_STRIDE + c] = 0.0f;

  unsigned gb0 = 0u, gb1 = 0u, gb2 = 0u;      // per-class gray bitmask (80 bits)
  float bestFilt = 0.0f;                      // > 0 iff some green && !gray gt seen
  float bd0 = 0.f, bd1 = 0.f, bd2 = 0.f, bd3 = 0.f;
  float minScore = 3.4e38f;                   // fallback: min-score gt (stable argsort[0])
  float md0 = 0.f, md1 = 0.f, md2 = 0.f, md3 = 0.f;

  for (int g = 0; g < G_GT; ++g) {
    const float bx = s_bbox[4 * g + 0];
    const float by = s_bbox[4 * g + 1];
    const float bw = s_bbox[4 * g + 2];
    const float bh = s_bbox[4 * g + 3];
    const int  lab = s_label[g];

    const float cx = bx + 0.5f * bw;
    const float cy = by + 0.5f * bh;
    float xs = fabsf(px - cx) * inv_half; xs = fmaxf(xs, 1.0f);
    float ys = fabsf(py - cy) * inv_half; ys = fmaxf(ys, 1.0f);
    const float score = __builtin_sqrtf(1.0f / xs) * __builtin_sqrtf(1.0f / ys);

    const float d0 = (px - bx) * inv_up;
    const float d1 = (py - by) * inv_up;
    const float d2 = (bx + bw - 1.0f - px) * inv_up;
    const float d3 = (by + bh - 1.0f - py) * inv_up;
    const float mind = fminf(fminf(d0, d1), fminf(d2, d3));
    const bool  hit  = (mind >= 0.0f);

    const float larger = fmaxf(bw, bh);
    const bool green = (lo <= larger) && (larger <= up) && hit;
    const bool grayb = ((((glo <= larger) && (larger < lo)) ||
                         ((up  <  larger) && (larger <= gup)))) && hit;

    if (green) {  // per-class scatter-max (order independent, matches .at[].max)
      const int idx = tid * CLS_STRIDE + lab;
      s_cls[idx] = fmaxf(s_cls[idx], score);
    }

    const unsigned bit  = 1u << (lab & 31);
    const int      word = lab >> 5;
    if (grayb) {
      gb0 |= (word == 0) ? bit : 0u;
      gb1 |= (word == 1) ? bit : 0u;
      gb2 |= (word == 2) ? bit : 0u;
    }

    // argmax(filt) over ascending-sorted order == max green&&!gray score,
    // ties -> smallest original index (strict >).
    const float filt = (green && !grayb) ? score : 0.0f;
    if (filt > bestFilt) { bestFilt = filt; bd0 = d0; bd1 = d1; bd2 = d2; bd3 = d3; }
    // fallback when filt all zero: sorted[0] == global min score, stable ties (strict <)
    if (score < minScore) { minScore = score; md0 = d0; md1 = d1; md2 = d2; md3 = d3; }
  }

  const bool haveBest = bestFilt > 0.0f;
  const float r0 = haveBest ? bd0 : md0;
  const float r1 = haveBest ? bd1 : md1;
  const float r2 = haveBest ? bd2 : md2;
  const float r3 = haveBest ? bd3 : md3;

  float* orow = out + ((size_t)b * P_TOTAL + p) * (NCLS + 4);  // 336 B rows, 16B aligned

#define GRAYBIT(cc) \
  (((((cc) < 32) ? gb0 : (((cc) < 64) ? gb1 : gb2)) >> ((cc) & 31)) & 1u)

#pragma unroll
  for (int c = 0; c < NCLS; c += 4) {
    float4 v;
    v.x = GRAYBIT(c + 0) ? -1.0f : s_cls[tid * CLS_STRIDE + c + 0];
    v.y = GRAYBIT(c + 1) ? -1.0f : s_cls[tid * CLS_STRIDE + c + 1];
    v.z = GRAYBIT(c + 2) ? -1.0f : s_cls[tid * CLS_STRIDE + c + 2];
    v.w = GRAYBIT(c + 3) ? -1.0f : s_cls[tid * CLS_STRIDE + c + 3];
    reinterpret_cast<float4*>(orow)[c >> 2] = v;
  }
#undef GRAYBIT

  float4 r; r.x = r0; r.y = r1; r.z = r2; r.w = r3;
  reinterpret_cast<float4*>(orow)[NCLS >> 2] = r;
}

extern "C" void kernel_launch(void* const* d_in, const int* in_sizes, int n_in,
                              void* d_out, int out_size, void* d_ws, size_t ws_size,
                              hipStream_t stream) {
  const float* gt_bboxes   = (const float*)d_in[0];
  const int*   gt_labels   = (const int*)  d_in[1];
  const float* points      = (const float*)d_in[2];
  const float* reg_ranges  = (const float*)d_in[3];
  const float* gray_ranges = (const float*)d_in[4];
  const float* strides     = (const float*)d_in[5];
  float* out = (float*)d_out;

  const int B = in_sizes[1] / G_GT;          // gt_labels is [B, G]
  dim3 grid(P_TOTAL / BLK, B);               // 21824 / 64 = 341 exactly
  LFD_assign_kernel<<<grid, BLK, 0, stream>>>(gt_bboxes, gt_labels, points,
                                              reg_ranges, gray_ranges, strides, out);
}